// Net_14147622273477
// MI455X (gfx1250) — compile-verified
//
#include <hip/hip_runtime.h>
#include <math.h>

#define T_ 8
#define N_ 50000
#define F_ 128
#define H_ 64
#define C_ 4
#define E_ 800000

typedef float v2f __attribute__((ext_vector_type(2)));
typedef float v8f __attribute__((ext_vector_type(8)));
typedef unsigned u32x4 __attribute__((ext_vector_type(4)));
typedef unsigned u32x8 __attribute__((ext_vector_type(8)));

// ---------------------------------------------------------------------------
// fill a float buffer with a constant
// ---------------------------------------------------------------------------
__global__ void fill_kernel(float* __restrict__ p, int n, float v) {
    int i = blockIdx.x * blockDim.x + threadIdx.x;
    if (i < n) p[i] = v;
}

// ---------------------------------------------------------------------------
// degree count: deg[dst] += 1 for every edge (deg pre-filled with 1.0 = self loop)
// ---------------------------------------------------------------------------
__global__ void deg_count_kernel(const int* __restrict__ dst_idx,
                                 float* __restrict__ deg) {
    int e = blockIdx.x * blockDim.x + threadIdx.x;
    if (e < E_) atomicAdd(&deg[dst_idx[e]], 1.0f);
}

// in-place deg -> rsqrt(deg)  (deg >= 1 always due to self loop)
__global__ void rsqrt_kernel(float* __restrict__ d, int n) {
    int i = blockIdx.x * blockDim.x + threadIdx.x;
    if (i < n) d[i] = rsqrtf(d[i]);
}

// ---------------------------------------------------------------------------
// xw = x_t @ Wg_t   [N,128] @ [128,64] -> [N,64], f32 WMMA 16x16x4
// block = 256 threads = 8 waves; each wave owns one 16x16 output tile.
// Wg_t (32 KB) is staged into LDS by the Tensor Data Mover: wave 0 issues a
// 1-D tensor_load_to_lds (D# per CDNA5 ISA), waits on
// TENSORcnt, and the workgroup barrier releases the other waves.
// N_ % 16 == 0  =>  every active wave's tile is fully in-bounds (no masking).
// ISA layouts:
//   A 16x4 f32 : vgpr v holds K = 2*(lane>>4) + v, M = lane&15
//   B 4x16 f32 : lane = 16*(K>>1) + n ; vgpr = K&1   (mirrors A)
//   C/D 16x16  : M = vgpr + 8*(lane>>4), N = lane&15
// ---------------------------------------------------------------------------
__global__ __launch_bounds__(256) void gemm_xw_wmma(
    const float* __restrict__ x,   // [N,128]  (timestep slice)
    const float* __restrict__ Wg,  // [128,64] (timestep slice)
    float* __restrict__ xw)        // [N,64]
{
    __shared__ float Wl[F_ * H_];              // 32 KB of the 320 KB WGP LDS
    const int tid = threadIdx.x;

    if (tid < 32) {
        // --- TDM descriptor: 1-D tensor, 8192 x f32, global -> LDS --------
        unsigned lds_off = (unsigned)(size_t)(__attribute__((address_space(3))) float*)Wl;
        unsigned long long ga = (unsigned long long)(const void*)Wg;
        // group0: [1:0]count=1 | [63:32]lds_addr | [120:64]global_addr | [127:126]type=2
        u32x4 g0 = { 1u, lds_off, (unsigned)ga,
                     (unsigned)(ga >> 32) | 0x80000000u };
        // group1: data_size=2(4B); tensor_dim0=tile_dim0=stride0=8192;
        //         tensor_dim1=tile_dim1=1; stride1=8192; no pad/iter/multicast
        u32x8 g1 = { 0x00020000u,   // workgroup_mask=0, data_size=2
                     0x20000000u,   // tensor_dim0[15:0]=0x2000 in [31:16]
                     0x00010000u,   // tensor_dim0[31:16]=0, tensor_dim1[15:0]=1
                     0x20000000u,   // tensor_dim1[31:16]=0, tile_dim0=0x2000
                     0x00000001u,   // tile_dim1=1, tile_dim2=0
                     0x00002000u,   // tensor_dim0_stride[31:0]=8192
                     0x20000000u,   // stride0[47:32]=0, stride1[15:0]=0x2000
                     0x00000000u }; // stride1[47:16]=0
        asm volatile("tensor_load_to_lds %0, %1" :: "s"(g0), "s"(g1) : "memory");
        __builtin_amdgcn_s_wait_tensorcnt(0);
    }
    __syncthreads();

    const int wave    = tid >> 5;
    const int lane    = tid & 31;
    const int half    = lane >> 4;             // 0/1
    const int lanelo  = lane & 15;
    const int colBase = (wave & 3) * 16;
    const int rowBase = (blockIdx.x * 2 + (wave >> 2)) * 16;
    if (rowBase >= N_) return;                 // uniform per wave; else tile fully valid

    const float* xr = x + (size_t)(rowBase + lanelo) * F_;

    v8f c = {};
#pragma unroll 8
    for (int k = 0; k < F_; k += 4) {
        const int kk = k + 2 * half;           // K index held by vgpr 0
        v2f a, b;
        a.x = xr[kk];
        a.y = xr[kk + 1];
        b.x = Wl[kk * H_ + colBase + lanelo];
        b.y = Wl[(kk + 1) * H_ + colBase + lanelo];
        c = __builtin_amdgcn_wmma_f32_16x16x4_f32(
                /*neg_a=*/false, a, /*neg_b=*/false, b,
                /*c_mod=*/(short)0, c, /*reuse_a=*/false, /*reuse_b=*/false);
    }

    float* outp = xw + (size_t)(rowBase + 8 * half) * H_ + colBase + lanelo;
#pragma unroll
    for (int v = 0; v < 8; ++v) outp[(size_t)v * H_] = c[v];
}

// ---------------------------------------------------------------------------
// edge scatter: agg[dst] += xw[src] * dinv[src]*dinv[dst]
// one wave32 per edge; lane handles a float2 of the 64-wide feature row
// ---------------------------------------------------------------------------
__global__ __launch_bounds__(256) void scatter_kernel(
    const int* __restrict__ src_idx, const int* __restrict__ dst_idx,
    const float* __restrict__ xw, const float* __restrict__ dinv,
    float* __restrict__ agg)
{
    const int wave = threadIdx.x >> 5;
    const int lane = threadIdx.x & 31;
    const int e = blockIdx.x * 8 + wave;
    if (e >= E_) return;                       // uniform per wave
    const int s = src_idx[e];
    const int d = dst_idx[e];
    const float norm = dinv[s] * dinv[d];
    const float2 v = ((const float2*)(xw + (size_t)s * H_))[lane];
    float* ag = agg + (size_t)d * H_ + 2 * lane;
    atomicAdd(ag,     v.x * norm);
    atomicAdd(ag + 1, v.y * norm);
}

// ---------------------------------------------------------------------------
// per-timestep finalize: acc += relu(agg + xw*dinv^2 (self loop) + bias),
// and reset agg to 0 for the next timestep (replaces a separate fill pass)
// ---------------------------------------------------------------------------
__global__ void finalize_kernel(float* __restrict__ agg,
                                const float* __restrict__ xw,
                                const float* __restrict__ dinv,
                                const float* __restrict__ bg_t,
                                float* __restrict__ acc)
{
    const int i = blockIdx.x * blockDim.x + threadIdx.x;
    if (i >= N_ * H_) return;
    const int node = i >> 6;
    const int h    = i & (H_ - 1);
    const float di = dinv[node];
    const float v  = agg[i] + xw[i] * di * di + bg_t[h];
    agg[i] = 0.0f;                             // ready for next timestep
    acc[i] += fmaxf(v, 0.0f);
}

// ---------------------------------------------------------------------------
// MLP head + log_softmax. One thread per node, weights staged in LDS.
// ---------------------------------------------------------------------------
__global__ __launch_bounds__(256) void mlp_head_kernel(
    const float* __restrict__ acc,
    const float* __restrict__ W1, const float* __restrict__ b1,
    const float* __restrict__ W2, const float* __restrict__ b2,
    const float* __restrict__ W3, const float* __restrict__ b3,
    float* __restrict__ out)
{
    __shared__ float sW1[H_ * 32], sW2[32 * 16], sW3[16 * C_];
    __shared__ float sb1[32], sb2[16], sb3[C_];
    const int tid = threadIdx.x;
    for (int i = tid; i < H_ * 32; i += 256) sW1[i] = W1[i];
    for (int i = tid; i < 32 * 16; i += 256) sW2[i] = W2[i];
    for (int i = tid; i < 16 * C_; i += 256) sW3[i] = W3[i];
    if (tid < 32) sb1[tid] = b1[tid];
    if (tid < 16) sb2[tid] = b2[tid];
    if (tid < C_) sb3[tid] = b3[tid];
    __syncthreads();

    const int node = blockIdx.x * 256 + tid;
    if (node >= N_) return;

    float h0[H_];
    const float* a = acc + (size_t)node * H_;
#pragma unroll
    for (int i = 0; i < H_; ++i) h0[i] = fmaxf(a[i], 0.0f);

    float h1[32];
    for (int j = 0; j < 32; ++j) {
        float s = sb1[j];
        for (int k = 0; k < H_; ++k) s = fmaf(h0[k], sW1[k * 32 + j], s);
        h1[j] = fmaxf(s, 0.0f);
    }
    float h2[16];
    for (int j = 0; j < 16; ++j) {
        float s = sb2[j];
        for (int k = 0; k < 32; ++k) s = fmaf(h1[k], sW2[k * 16 + j], s);
        h2[j] = fmaxf(s, 0.0f);
    }
    float lg[C_];
    float m = -1e30f;
    for (int j = 0; j < C_; ++j) {
        float s = sb3[j];
        for (int k = 0; k < 16; ++k) s = fmaf(h2[k], sW3[k * C_ + j], s);
        lg[j] = s;
        m = fmaxf(m, s);
    }
    float se = 0.0f;
    for (int j = 0; j < C_; ++j) se += __expf(lg[j] - m);
    const float lse = __logf(se);
    for (int j = 0; j < C_; ++j) out[(size_t)node * C_ + j] = lg[j] - m - lse;
}

// ---------------------------------------------------------------------------
// host-side orchestration (graph-capture safe: only kernel launches on stream)
// ---------------------------------------------------------------------------
extern "C" void kernel_launch(void* const* d_in, const int* in_sizes, int n_in,
                              void* d_out, int out_size, void* d_ws, size_t ws_size,
                              hipStream_t stream) {
    const float* x     = (const float*)d_in[0];  // [T,N,F]
    const int*   edges = (const int*)  d_in[1];  // [T,2,E]
    const float* Wg    = (const float*)d_in[2];  // [T,F,H]
    const float* bg    = (const float*)d_in[3];  // [T,H]
    const float* W1    = (const float*)d_in[4];
    const float* b1    = (const float*)d_in[5];
    const float* W2    = (const float*)d_in[6];
    const float* b2    = (const float*)d_in[7];
    const float* W3    = (const float*)d_in[8];
    const float* b3    = (const float*)d_in[9];
    float* out = (float*)d_out;

    float* ws   = (float*)d_ws;
    float* acc  = ws;                           // N*H
    float* xw   = ws + (size_t)N_ * H_;         // N*H
    float* agg  = ws + 2 * (size_t)N_ * H_;     // N*H
    float* dinv = ws + 3 * (size_t)N_ * H_;     // N

    const int NH = N_ * H_;
    fill_kernel<<<(NH + 255) / 256, 256, 0, stream>>>(acc, NH, 0.0f);
    fill_kernel<<<(NH + 255) / 256, 256, 0, stream>>>(agg, NH, 0.0f);

    for (int t = 0; t < T_; ++t) {
        const float* x_t  = x  + (size_t)t * N_ * F_;
        const float* Wg_t = Wg + (size_t)t * F_ * H_;
        const float* bg_t = bg + (size_t)t * H_;
        const int* src_t  = edges + (size_t)t * 2 * E_;
        const int* dst_t  = src_t + E_;

        // degrees -> dinv
        fill_kernel<<<(N_ + 255) / 256, 256, 0, stream>>>(dinv, N_, 1.0f);
        deg_count_kernel<<<(E_ + 255) / 256, 256, 0, stream>>>(dst_t, dinv);
        rsqrt_kernel<<<(N_ + 255) / 256, 256, 0, stream>>>(dinv, N_);

        // dense projection via f32 WMMA (Wg staged by TDM)
        gemm_xw_wmma<<<(N_ + 31) / 32, 256, 0, stream>>>(x_t, Wg_t, xw);

        // normalized aggregation (agg zeroed by previous finalize / initial fill)
        scatter_kernel<<<(E_ + 7) / 8, 256, 0, stream>>>(src_t, dst_t, xw, dinv, agg);

        // self-loop + bias + relu + temporal accumulate + agg reset
        finalize_kernel<<<(NH + 255) / 256, 256, 0, stream>>>(agg, xw, dinv, bg_t, acc);
    }

    mlp_head_kernel<<<(N_ + 255) / 256, 256, 0, stream>>>(
        acc, W1, b1, W2, b2, W3, b3, out);
}